// GraphSAGE_6837587935744
// MI455X (gfx1250) — compile-verified
//
#include <hip/hip_runtime.h>
#include <stdint.h>

// GraphSAGE forward on MI455X (gfx1250, wave32).
//
// ws layout (floats): [deg N][agg1 N*128][agg2 N*512][h N*512]
// deg/agg1/agg2 are zero-initialized each call (contiguous span), h is fully
// overwritten by the WMMA GEMM.

#define NN 50000
#define NE 600000
#define IN_DIM 128
#define HID_DIM 512
#define ASTR 132   // LDS row stride (floats): 16B-aligned, 4-bank skew per row

typedef float v2f __attribute__((ext_vector_type(2)));
typedef float v8f __attribute__((ext_vector_type(8)));

// ---------------------------------------------------------------- zero init
__global__ void k_zero(float* __restrict__ p, long long n) {
    long long i = (long long)blockIdx.x * blockDim.x + threadIdx.x;
    long long stride = (long long)gridDim.x * blockDim.x;
    for (; i < n; i += stride) p[i] = 0.0f;
}

// ---------------------------------------------------------------- degrees
__global__ void k_degree(const int* __restrict__ dst, float* __restrict__ deg) {
    int e = blockIdx.x * blockDim.x + threadIdx.x;
    if (e < NE) atomicAdd(&deg[dst[e]], 1.0f);
}

// ------------------------------------------------- layer-1 scatter (128-d)
__global__ void k_scatter1(const float* __restrict__ x,
                           const int* __restrict__ src,
                           const int* __restrict__ dst,
                           float* __restrict__ agg1) {
    long long total = (long long)NE * IN_DIM;
    long long i = (long long)blockIdx.x * blockDim.x + threadIdx.x;
    long long stride = (long long)gridDim.x * blockDim.x;
    for (; i < total; i += stride) {
        int e = (int)(i >> 7);        // /128
        int c = (int)(i & 127);
        atomicAdd(&agg1[(long long)dst[e] * IN_DIM + c],
                  x[(long long)src[e] * IN_DIM + c]);
    }
}

// ----------------------------------------- layer-1 GEMM via f32 WMMA + ReLU
// h = relu( (agg1 * invdeg) @ W1l + x @ W1r + b1 )
// A tiles (16 rows of x and agg1) staged once per block into LDS with the
// CDNA5 async copy engine (global_load_async_to_lds_b128 -> ASYNCcnt), then
// 8 waves each compute one 16x16 tile from LDS fragments. Two accumulators
// keep two independent WMMA chains in flight.
// grid = (HID_DIM/16/8, NN/16) = (4, 3125), block = 256.
__global__ void k_gemm1(const float* __restrict__ x,
                        const float* __restrict__ agg1,
                        const float* __restrict__ deg,
                        const float* __restrict__ W1l,
                        const float* __restrict__ W1r,
                        const float* __restrict__ b1,
                        float* __restrict__ h) {
    __shared__ float sX[16 * ASTR];   // x tile,    row-major, padded
    __shared__ float sG[16 * ASTR];   // agg1 tile, row-major, padded

    const int lane = threadIdx.x & 31;
    const int wave = threadIdx.x >> 5;
    const int nt = blockIdx.x * 8 + wave;   // 0..31 column tile
    const int mt = blockIdx.y;              // 0..3124 row tile

    // ---- async-stage the two A tiles: 32 row-copies, 4 per wave ----------
    // one instruction moves a full 128-float row (32 lanes x b128)
    for (int t = wave; t < 32; t += 8) {
        const int rrow = t & 15;
        const float* gsrc = (t < 16 ? x : agg1)
                          + (long long)(mt * 16 + rrow) * IN_DIM + lane * 4;
        float* ldst = (t < 16 ? sX : sG) + rrow * ASTR + lane * 4;
        uint32_t loff = (uint32_t)(uintptr_t)ldst;   // wave-relative LDS byte addr
        asm volatile("global_load_async_to_lds_b128 %0, %1, off"
                     :: "v"(loff), "v"(gsrc) : "memory");
    }
    asm volatile("s_wait_asynccnt 0x0" ::: "memory");
    __syncthreads();

    const int r  = lane & 15;               // A row within tile / C column
    const int kg = lane >> 4;               // K-pair group (0 or 1)
    const int n  = lane & 15;               // C/D + B column within tile
    const int row = mt * 16 + r;

    const float invd = 1.0f / fmaxf(deg[row], 1.0f);

    // two accumulators: bias pre-loaded into cx, ca starts at zero
    const float bias = b1[nt * 16 + n];
    v8f cx, ca;
#pragma unroll
    for (int v = 0; v < 8; ++v) { cx[v] = bias; ca[v] = 0.0f; }

    const float* __restrict__ Bl = W1l + nt * 16;   // W1l[k][n] = Bl[k*HID + n]
    const float* __restrict__ Br = W1r + nt * 16;
    const float* __restrict__ ax = sX + r * ASTR;   // LDS fragments
    const float* __restrict__ ag = sG + r * ASTR;

    for (int k = 0; k < IN_DIM; k += 4) {
        const int ka = k + kg * 2;          // this lane's K pair (even -> 8B aligned)
        v2f a, b;

        // ---- x @ W1r  (chain 1) ----
        a = *(const v2f*)&ax[ka];                       // ds_load_b64, bank-clean
        b[0] = Br[(long long)ka * HID_DIM + n];
        b[1] = Br[(long long)(ka + 1) * HID_DIM + n];
        cx = __builtin_amdgcn_wmma_f32_16x16x4_f32(false, a, false, b,
                                                   (short)0, cx, false, false);

        // ---- (agg1 * invdeg) @ W1l  (chain 2) ----
        a = *(const v2f*)&ag[ka];
        a[0] *= invd;
        a[1] *= invd;
        b[0] = Bl[(long long)ka * HID_DIM + n];
        b[1] = Bl[(long long)(ka + 1) * HID_DIM + n];
        ca = __builtin_amdgcn_wmma_f32_16x16x4_f32(false, a, false, b,
                                                   (short)0, ca, false, false);
    }

    // D layout: lane holds column n, rows (kg*8 + v)
    const int mb = kg * 8;
#pragma unroll
    for (int v = 0; v < 8; ++v) {
        float val = fmaxf(cx[v] + ca[v], 0.0f);         // merge chains + ReLU
        h[(long long)(mt * 16 + mb + v) * HID_DIM + nt * 16 + n] = val;
    }
}

// ------------------------------------------------- layer-2 scatter (512-d)
// agg2 (102 MB) fits in the 192 MB L2: f32 atomics resolve at L2 bandwidth.
__global__ void k_scatter2(const float* __restrict__ h,
                           const int* __restrict__ src,
                           const int* __restrict__ dst,
                           float* __restrict__ agg2) {
    long long total = (long long)NE * HID_DIM;
    long long i = (long long)blockIdx.x * blockDim.x + threadIdx.x;
    long long stride = (long long)gridDim.x * blockDim.x;
    for (; i < total; i += stride) {
        int e = (int)(i >> 9);        // /512
        int c = (int)(i & 511);
        atomicAdd(&agg2[(long long)dst[e] * HID_DIM + c],
                  h[(long long)src[e] * HID_DIM + c]);
    }
}

// ------------------------------------- layer-2 (512 -> 2) + log_softmax
// One wave32 per node; shuffle reduction.
__global__ void k_out(const float* __restrict__ h,
                      const float* __restrict__ agg2,
                      const float* __restrict__ deg,
                      const float* __restrict__ W2l,
                      const float* __restrict__ b2,
                      const float* __restrict__ W2r,
                      float* __restrict__ out) {
    const int lane = threadIdx.x & 31;
    const int wave = threadIdx.x >> 5;
    const int node = blockIdx.x * 8 + wave;
    if (node >= NN) return;

    const float invd = 1.0f / fmaxf(deg[node], 1.0f);
    const float* __restrict__ hn = h    + (long long)node * HID_DIM;
    const float* __restrict__ an = agg2 + (long long)node * HID_DIM;

    float a0 = 0.0f, a1 = 0.0f;
    for (int c = lane; c < HID_DIM; c += 32) {
        float hv = hn[c];
        float av = an[c] * invd;
        a0 = fmaf(av, W2l[c * 2 + 0], fmaf(hv, W2r[c * 2 + 0], a0));
        a1 = fmaf(av, W2l[c * 2 + 1], fmaf(hv, W2r[c * 2 + 1], a1));
    }
#pragma unroll
    for (int off = 16; off > 0; off >>= 1) {
        a0 += __shfl_xor(a0, off, 32);
        a1 += __shfl_xor(a1, off, 32);
    }
    if (lane == 0) {
        float o0 = a0 + b2[0];
        float o1 = a1 + b2[1];
        float m = fmaxf(o0, o1);
        float lse = m + __logf(__expf(o0 - m) + __expf(o1 - m));
        out[(long long)node * 2 + 0] = o0 - lse;
        out[(long long)node * 2 + 1] = o1 - lse;
    }
}

// ---------------------------------------------------------------- launcher
extern "C" void kernel_launch(void* const* d_in, const int* in_sizes, int n_in,
                              void* d_out, int out_size, void* d_ws, size_t ws_size,
                              hipStream_t stream) {
    const float* x   = (const float*)d_in[0];
    const float* W1l = (const float*)d_in[1];
    const float* b1  = (const float*)d_in[2];
    const float* W1r = (const float*)d_in[3];
    const float* W2l = (const float*)d_in[4];
    const float* b2  = (const float*)d_in[5];
    const float* W2r = (const float*)d_in[6];
    const int*   ei  = (const int*)d_in[7];
    const int* src = ei;              // edge_index[0]
    const int* dst = ei + NE;         // edge_index[1]
    float* out = (float*)d_out;

    float* ws   = (float*)d_ws;
    float* deg  = ws;                                   // N
    float* agg1 = deg  + NN;                            // N*128
    float* agg2 = agg1 + (long long)NN * IN_DIM;        // N*512
    float* h    = agg2 + (long long)NN * HID_DIM;       // N*512 (no zeroing needed)

    // zero deg + agg1 + agg2 (one contiguous span)
    long long zn = (long long)NN * (1 + IN_DIM + HID_DIM);
    k_zero<<<4096, 256, 0, stream>>>(deg, zn);

    // degrees
    k_degree<<<(NE + 255) / 256, 256, 0, stream>>>(dst, deg);

    // layer-1 aggregation
    {
        long long total = (long long)NE * IN_DIM;
        int blocks = (int)((total + 255) / 256);
        k_scatter1<<<blocks, 256, 0, stream>>>(x, src, dst, agg1);
    }

    // layer-1 fused dual GEMM + bias + ReLU (f32 WMMA, async-LDS A tiles)
    {
        dim3 grid(HID_DIM / 16 / 8, NN / 16);   // (4, 3125)
        k_gemm1<<<grid, 256, 0, stream>>>(x, agg1, deg, W1l, W1r, b1, h);
    }

    // layer-2 aggregation
    {
        long long total = (long long)NE * HID_DIM;
        int blocks = (int)((total + 255) / 256);
        k_scatter2<<<blocks, 256, 0, stream>>>(h, src, dst, agg2);
    }

    // layer-2 projection + log_softmax
    k_out<<<(NN + 7) / 8, 256, 0, stream>>>(h, agg2, deg, W2l, b2, W2r, out);
}